// AttentionAggregate_Cos_59751585022699
// MI455X (gfx1250) — compile-verified
//
#include <hip/hip_runtime.h>
#include <cstddef>

typedef __attribute__((ext_vector_type(4))) float vf4;
typedef __attribute__((ext_vector_type(2))) float v2f;
typedef __attribute__((ext_vector_type(8))) float v8f;

#define KNB 32
#define DF  128
#define EPSV 1e-8f

__global__ __launch_bounds__(256)
void cosattn_wmma_kernel(const float* __restrict__ middle_key,
                         const float* __restrict__ nodes_key,
                         const float* __restrict__ middle_value,
                         float* __restrict__ out, int N)
{
    const int lane = threadIdx.x & 31;
    const int n = blockIdx.x * 8 + (threadIdx.x >> 5);
    if (n >= N) return;  // wave-uniform branch

    // ---- Stage 1: lane k owns neighbor k. dot + norms over D=128 ----
    const vf4* kq  = reinterpret_cast<const vf4*>(nodes_key + (size_t)n * DF);
    const vf4* mkp = reinterpret_cast<const vf4*>(middle_key +
                        ((size_t)n * KNB + lane) * DF);
    float accd = 0.f, accm = 0.f, accn = 0.f;
#pragma unroll 8
    for (int c = 0; c < DF / 4; ++c) {
        vf4 kv = kq[c];                                  // wave-uniform row
        vf4 mk = __builtin_nontemporal_load(mkp + c);    // streaming, no reuse
        accd += mk[0]*kv[0] + mk[1]*kv[1] + mk[2]*kv[2] + mk[3]*kv[3];
        accm += mk[0]*mk[0] + mk[1]*mk[1] + mk[2]*mk[2] + mk[3]*mk[3];
        accn += kv[0]*kv[0] + kv[1]*kv[1] + kv[2]*kv[2] + kv[3]*kv[3];
    }
    float sim = accd / fmaxf(sqrtf(accn) * sqrtf(accm), EPSV);
    float t = tanhf(sim);

    // ---- softmax over K=32 == full wave32 reduction ----
    float m = t;
    for (int off = 16; off > 0; off >>= 1)
        m = fmaxf(m, __shfl_xor(m, off, 32));
    float e = __expf(t - m);
    float s = e;
    for (int off = 16; off > 0; off >>= 1)
        s += __shfl_xor(s, off, 32);
    float w = e / s;                      // lane k holds softmax weight[k]

    // ---- Stage 2: out[d] = sum_k w[k]*mv[k,d] via V_WMMA_F32_16X16X4_F32 ----
    // A(16x4): all 16 rows = weight chunk {w[k0..k0+3]}; ISA layout:
    //   VGPR0 = K=0 (lanes 0-15) / K=2 (lanes 16-31), VGPR1 = K=1 / K=3.
    // B(4x16): natural mv tile; VGPR0 = row K=0 / K=2, VGPR1 = row K=1 / K=3.
    // => every row of D equals the 16-wide output chunk (accumulated over C).
    const int half = lane >> 4;
    const int col  = lane & 15;
    v2f a[8];
#pragma unroll
    for (int c = 0; c < 8; ++c) {
        a[c][0] = __shfl(w, 4 * c + 2 * half,     32);
        a[c][1] = __shfl(w, 4 * c + 2 * half + 1, 32);
    }
    const float* mvb = middle_value + (size_t)n * KNB * DF;
    float* op = out + (size_t)n * DF;
#pragma unroll 2
    for (int tl = 0; tl < 8; ++tl) {
        const int d0 = tl * 16;
        v8f acc = {};
#pragma unroll
        for (int c = 0; c < 8; ++c) {
            const int kr = 4 * c + 2 * half;
            v2f b;
            b[0] = __builtin_nontemporal_load(mvb + (size_t)kr * DF + d0 + col);
            b[1] = __builtin_nontemporal_load(mvb + (size_t)(kr + 1) * DF + d0 + col);
            acc = __builtin_amdgcn_wmma_f32_16x16x4_f32(
                /*neg_a=*/false, a[c], /*neg_b=*/false, b,
                /*c_mod=*/(short)0, acc, /*reuse_a=*/false, /*reuse_b=*/false);
        }
        if (lane < 16) op[d0 + col] = acc[0];  // rows identical; N = col
    }
}

extern "C" void kernel_launch(void* const* d_in, const int* in_sizes, int n_in,
                              void* d_out, int out_size, void* d_ws, size_t ws_size,
                              hipStream_t stream) {
    const float* middle_key   = (const float*)d_in[0];
    const float* nodes_key    = (const float*)d_in[1];
    const float* middle_value = (const float*)d_in[2];
    float* out = (float*)d_out;

    const int N = in_sizes[1] / DF;          // nodes_key is [N, 128]
    const int waves_per_block = 8;           // 256 threads = 8 wave32
    dim3 grid((N + waves_per_block - 1) / waves_per_block);
    dim3 block(32 * waves_per_block);
    cosattn_wmma_kernel<<<grid, block, 0, stream>>>(middle_key, nodes_key,
                                                    middle_value, out, N);
}